// TRAN_NER_16776142258841
// MI455X (gfx1250) — compile-verified
//
#include <hip/hip_runtime.h>
#include <hip/hip_bf16.h>
#include <math.h>

// ---------------------------------------------------------------------------
// Transition-based NER parser forward pass for MI455X (gfx1250, wave32).
//
// Phase 1 (parallel, matrix cores): gather buf_in = [buffer_start; rev word
//   embeds] (257 x 300) and compute preBuf = Wih_buffer @ buf_in^T with
//   v_wmma_f32_16x16x4_f32 (fp32 WMMA -> exact numerics; workload is
//   latency-bound so lower-precision matrix throughput buys nothing).
// Phase 2 (serial, one persistent 32-wave block): buffer LSTM recurrence
//   (Whh@h matvec per step), then the action loop with stack/output/
//   action/entity LSTMs, combiner, scorer, subset log-softmax.
// ---------------------------------------------------------------------------

typedef float v2f __attribute__((ext_vector_type(2)));
typedef float v8f __attribute__((ext_vector_type(8)));

#define E_DIM 300
#define H_DIM 512
#define AD_DIM 64
#define RD_DIM 64
#define ASZ_DIM 6
#define G_DIM (4 * H_DIM)   // 2048
#define EG_DIM (4 * E_DIM)  // 1200

__device__ __forceinline__ float sigm(float x) { return 1.0f / (1.0f + expf(-x)); }

// ------------------------------ gather -------------------------------------
__global__ void k_gather_buf_in(const int* __restrict__ sent,
                                const float* __restrict__ word_embeds,
                                const float* __restrict__ buffer_start,
                                float* __restrict__ buf_in, int L) {
  int r = blockIdx.x;  // 0..L ; row 0 = buffer_start, row r = embeds[sent[L-r]]
  const float* src =
      (r == 0) ? buffer_start : (word_embeds + (size_t)sent[L - r] * E_DIM);
  for (int j = threadIdx.x; j < E_DIM; j += blockDim.x)
    buf_in[(size_t)r * E_DIM + j] = src[j];
}

// --------------------------- WMMA f32 GEMM ---------------------------------
// D[n][m] = sum_k W[m][k] * X[n][k].  W: MxK row-major, X: NxK row-major,
// D: NxM row-major.  One wave per 16(M)x16(N) tile; K stepped by 4.
// A frag (16x4): lanes 0-15 -> K={kk,kk+1}, lanes 16-31 -> K={kk+2,kk+3},
// M = lane%16.  B frag mirrors A with N = lane%16.
// C/D: lane = N + 16*(M>=8), vgpr v -> M = 8*(lane/16)+v  => each lane stores
// 8 contiguous M elements of row N.
//
// NOTE: lanes with n >= N load unconditionally (addresses stay inside our own
// workspace allocation just past buf_in); their garbage only feeds D columns
// that are discarded at the guarded store.  This keeps EXEC untouched so A and
// B loads both become plain global_load_b64 with no divergent branches.
__global__ __launch_bounds__(32) void k_wmma_gemm_wxT(
    const float* __restrict__ W, const float* __restrict__ X,
    float* __restrict__ D, int M, int K, int N) {
  int tilesM = M >> 4;
  int tm = (blockIdx.x % tilesM) << 4;
  int tn = (blockIdx.x / tilesM) << 4;
  int lane = threadIdx.x & 31;
  int half = lane >> 4, l16 = lane & 15;
  int n = tn + l16;
  const float* Wr = W + (size_t)(tm + l16) * K + 2 * half;  // 8B aligned
  const float* Xr = X + (size_t)n * K + 2 * half;           // 8B aligned
  v8f c = {};
  for (int k = 0; k < K; k += 4) {
    v2f a = *(const v2f*)(Wr + k);
    v2f b = *(const v2f*)(Xr + k);
    c = __builtin_amdgcn_wmma_f32_16x16x4_f32(false, a, false, b, (short)0, c,
                                              false, false);
  }
  if (n < N) {
    float* out = D + (size_t)n * M + tm + 8 * half;
#pragma unroll
    for (int v = 0; v < 8; ++v) out[v] = c[v];
  }
}

// ------------------------- persistent serial phase -------------------------
#define NT 1024

__shared__ float sx[G_DIM];    // generic input / feature vector
__shared__ float sg[G_DIM];    // gate pre-activations / temp
__shared__ float sy[H_DIM];    // hlin temp
__shared__ float s_bh[H_DIM], s_bc[H_DIM];    // buffer LSTM state
__shared__ float s_oh[H_DIM], s_oc[H_DIM];    // output LSTM state
__shared__ float s_ah[H_DIM], s_ac[H_DIM];    // action LSTM state
__shared__ float s_sh[H_DIM], s_sc[H_DIM];    // stack-top LSTM state
__shared__ float s_fh[E_DIM], s_fc[E_DIM];    // ent_f state
__shared__ float s_gh[E_DIM], s_gc[E_DIM];    // ent_b state
__shared__ float s_scores[8];

// one full LSTM step: g = Wih@x + Whh@h + bih + bhh ; gates ; updates h,c
__device__ void lstm_step_dev(const float* __restrict__ Wih, int din,
                              const float* __restrict__ Whh,
                              const float* __restrict__ bih,
                              const float* __restrict__ bhh,
                              const float* __restrict__ x, float* h, float* c,
                              int hid) {
  int tid = threadIdx.x;
  int M = 4 * hid;
  for (int j = tid; j < M; j += NT) {
    float acc = bih[j] + bhh[j];
    const float* wi = Wih + (size_t)j * din;
    for (int k = 0; k < din; ++k) acc += wi[k] * x[k];
    const float* wh = Whh + (size_t)j * hid;
    for (int k = 0; k < hid; ++k) acc += wh[k] * h[k];
    sg[j] = acc;
  }
  __syncthreads();
  for (int j = tid; j < hid; j += NT) {
    float ig = sigm(sg[j]);
    float fg = sigm(sg[hid + j]);
    float gg = tanhf(sg[2 * hid + j]);
    float og = sigm(sg[3 * hid + j]);
    float cn = fg * c[j] + ig * gg;
    c[j] = cn;
    h[j] = og * tanhf(cn);
  }
  __syncthreads();
}

__global__ __launch_bounds__(NT) void k_parser_seq(
    const int* __restrict__ actions, int nAct, int L,
    const float* __restrict__ preBuf, const float* __restrict__ buf_in,
    float* __restrict__ buf_outs, float* __restrict__ stkE,
    float* __restrict__ stkH, float* __restrict__ stkC,
    const float* stack_Wih, const float* stack_Whh, const float* stack_bih,
    const float* stack_bhh, const float* buffer_Whh, const float* buffer_bih,
    const float* buffer_bhh, const float* output_Wih, const float* output_Whh,
    const float* output_bih, const float* output_bhh, const float* action_Wih,
    const float* action_Whh, const float* action_bih, const float* action_bhh,
    const float* entf_Wih, const float* entf_Whh, const float* entf_bih,
    const float* entf_bhh, const float* entb_Wih, const float* entb_Whh,
    const float* entb_bih, const float* entb_bhh, const float* comb_W,
    const float* comb_b, const float* act_W, const float* act_b,
    const float* er_W, const float* er_b, const float* stack_start,
    const float* output_start, const float* action_start,
    const float* action_embeds, const float* relation_embeds,
    float* __restrict__ losses) {
  int tid = threadIdx.x;

  // ---- 1. buffer LSTM over L+1 steps (x-projection precomputed via WMMA) --
  for (int j = tid; j < H_DIM; j += NT) { s_bh[j] = 0.0f; s_bc[j] = 0.0f; }
  __syncthreads();
  for (int t = 0; t <= L; ++t) {
    const float* pre = preBuf + (size_t)t * G_DIM;
    for (int j = tid; j < G_DIM; j += NT) {
      float acc = pre[j] + buffer_bih[j] + buffer_bhh[j];
      const float* wh = buffer_Whh + (size_t)j * H_DIM;
      for (int k = 0; k < H_DIM; ++k) acc += wh[k] * s_bh[k];
      sg[j] = acc;
    }
    __syncthreads();
    for (int j = tid; j < H_DIM; j += NT) {
      float ig = sigm(sg[j]);
      float fg = sigm(sg[H_DIM + j]);
      float gg = tanhf(sg[2 * H_DIM + j]);
      float og = sigm(sg[3 * H_DIM + j]);
      float cn = fg * s_bc[j] + ig * gg;
      s_bc[j] = cn;
      s_bh[j] = og * tanhf(cn);
    }
    __syncthreads();
    for (int j = tid; j < H_DIM; j += NT) buf_outs[(size_t)t * H_DIM + j] = s_bh[j];
    __syncthreads();
  }

  // ---- 2. initial stack/output/action steps -------------------------------
  for (int j = tid; j < E_DIM; j += NT) sx[j] = stack_start[j];
  for (int j = tid; j < H_DIM; j += NT) { s_sh[j] = 0.0f; s_sc[j] = 0.0f; }
  __syncthreads();
  lstm_step_dev(stack_Wih, E_DIM, stack_Whh, stack_bih, stack_bhh, sx, s_sh,
                s_sc, H_DIM);
  for (int j = tid; j < E_DIM; j += NT) stkE[j] = stack_start[j];
  for (int j = tid; j < H_DIM; j += NT) { stkH[j] = s_sh[j]; stkC[j] = s_sc[j]; }

  for (int j = tid; j < E_DIM; j += NT) sx[j] = output_start[j];
  for (int j = tid; j < H_DIM; j += NT) { s_oh[j] = 0.0f; s_oc[j] = 0.0f; }
  __syncthreads();
  lstm_step_dev(output_Wih, E_DIM, output_Whh, output_bih, output_bhh, sx,
                s_oh, s_oc, H_DIM);

  for (int j = tid; j < AD_DIM; j += NT) sx[j] = action_start[j];
  for (int j = tid; j < H_DIM; j += NT) { s_ah[j] = 0.0f; s_ac[j] = 0.0f; }
  __syncthreads();
  lstm_step_dev(action_Wih, AD_DIM, action_Whh, action_bih, action_bhh, sx,
                s_ah, s_ac, H_DIM);

  int buf_idx = L;  // L = sentence length
  int slen = 1;

  // ---- 3. action loop -----------------------------------------------------
  for (int t = 0; t < nAct; ++t) {
    int aid = actions[t];  // uniform across block

    // feat = [out_h, stack_top_h, buf_outs[buf_idx], act_h]
    const float* bo = buf_outs + (size_t)buf_idx * H_DIM;
    for (int j = tid; j < H_DIM; j += NT) {
      sx[j] = s_oh[j];
      sx[H_DIM + j] = s_sh[j];
      sx[2 * H_DIM + j] = bo[j];
      sx[3 * H_DIM + j] = s_ah[j];
    }
    __syncthreads();
    // hlin = tanh(comb_W @ feat + comb_b)
    for (int j = tid; j < H_DIM; j += NT) {
      float acc = comb_b[j];
      const float* wr = comb_W + (size_t)j * G_DIM;
      for (int k = 0; k < G_DIM; ++k) acc += wr[k] * sx[k];
      sy[j] = tanhf(acc);
    }
    __syncthreads();
    // scores = act_W @ hlin + act_b
    if (tid < ASZ_DIM) {
      float acc = act_b[tid];
      const float* wr = act_W + (size_t)tid * H_DIM;
      for (int k = 0; k < H_DIM; ++k) acc += wr[k] * sy[k];
      s_scores[tid] = acc;
    }
    __syncthreads();
    if (tid == 0) {
      bool hasBuf = (buf_idx >= 1);
      bool hasStk = (slen > 1);
      float mx = -3.4e38f;
      for (int v = 0; v < ASZ_DIM; ++v) {
        bool ok = (v < 2) ? hasBuf : hasStk;
        if (ok && s_scores[v] > mx) mx = s_scores[v];
      }
      float sum = 0.0f;
      for (int v = 0; v < ASZ_DIM; ++v) {
        bool ok = (v < 2) ? hasBuf : hasStk;
        if (ok) sum += expf(s_scores[v] - mx);
      }
      losses[t] = s_scores[aid] - mx - logf(sum);
    }
    __syncthreads();

    // action LSTM step with action_embeds[aid]
    for (int j = tid; j < AD_DIM; j += NT) sx[j] = action_embeds[aid * AD_DIM + j];
    __syncthreads();
    lstm_step_dev(action_Wih, AD_DIM, action_Whh, action_bih, action_bhh, sx,
                  s_ah, s_ac, H_DIM);

    if (aid == 0) {  // SHIFT
      const float* w = buf_in + (size_t)buf_idx * E_DIM;
      for (int j = tid; j < E_DIM; j += NT) sx[j] = w[j];
      __syncthreads();
      lstm_step_dev(stack_Wih, E_DIM, stack_Whh, stack_bih, stack_bhh, sx,
                    s_sh, s_sc, H_DIM);
      for (int j = tid; j < E_DIM; j += NT) stkE[(size_t)slen * E_DIM + j] = sx[j];
      for (int j = tid; j < H_DIM; j += NT) {
        stkH[(size_t)slen * H_DIM + j] = s_sh[j];
        stkC[(size_t)slen * H_DIM + j] = s_sc[j];
      }
      __syncthreads();
      ++slen;
      --buf_idx;
    } else if (aid == 1) {  // OUT
      const float* w = buf_in + (size_t)buf_idx * E_DIM;
      for (int j = tid; j < E_DIM; j += NT) sx[j] = w[j];
      __syncthreads();
      lstm_step_dev(output_Wih, E_DIM, output_Whh, output_bih, output_bhh, sx,
                    s_oh, s_oc, H_DIM);
      --buf_idx;
    } else {  // REDUCE: bidirectional entity LSTMs over stack contents
      for (int j = tid; j < E_DIM; j += NT) {
        s_fh[j] = 0.0f; s_fc[j] = 0.0f;
        s_gh[j] = 0.0f; s_gc[j] = 0.0f;
      }
      __syncthreads();
      for (int s = 0; s < slen; ++s) {
        const float* e = stkE + (size_t)s * E_DIM;
        for (int j = tid; j < E_DIM; j += NT) sx[j] = e[j];
        __syncthreads();
        lstm_step_dev(entf_Wih, E_DIM, entf_Whh, entf_bih, entf_bhh, sx, s_fh,
                      s_fc, E_DIM);
      }
      for (int s = slen - 1; s >= 0; --s) {
        const float* e = stkE + (size_t)s * E_DIM;
        for (int j = tid; j < E_DIM; j += NT) sx[j] = e[j];
        __syncthreads();
        lstm_step_dev(entb_Wih, E_DIM, entb_Whh, entb_bih, entb_bhh, sx, s_gh,
                      s_gc, E_DIM);
      }
      // ent = tanh(er_W @ [efwd; ebwd; r_emb] + er_b)
      for (int j = tid; j < E_DIM; j += NT) {
        sx[j] = s_fh[j];
        sx[E_DIM + j] = s_gh[j];
      }
      for (int j = tid; j < RD_DIM; j += NT)
        sx[2 * E_DIM + j] = relation_embeds[aid * RD_DIM + j];
      __syncthreads();
      const int KE = 2 * E_DIM + RD_DIM;  // 664
      for (int j = tid; j < E_DIM; j += NT) {
        float acc = er_b[j];
        const float* wr = er_W + (size_t)j * KE;
        for (int k = 0; k < KE; ++k) acc += wr[k] * sx[k];
        sg[j] = tanhf(acc);
      }
      __syncthreads();
      for (int j = tid; j < E_DIM; j += NT) sx[j] = sg[j];
      __syncthreads();
      lstm_step_dev(output_Wih, E_DIM, output_Whh, output_bih, output_bhh, sx,
                    s_oh, s_oc, H_DIM);
      // stack reset to its first element; restore top state from slot 0
      for (int j = tid; j < H_DIM; j += NT) { s_sh[j] = stkH[j]; s_sc[j] = stkC[j]; }
      __syncthreads();
      slen = 1;
    }
  }
}

// ------------------------------ launch -------------------------------------
extern "C" void kernel_launch(void* const* d_in, const int* in_sizes, int n_in,
                              void* d_out, int out_size, void* d_ws,
                              size_t ws_size, hipStream_t stream) {
  // Input order assumption: setup_inputs() dict order, params in insertion
  // order (see reference _params()).
  int i = 0;
  const int* sentence = (const int*)d_in[i++];        // (L,1) int32
  const int* actions = (const int*)d_in[i++];         // (nAct,1) int32
  const float* word_embeds = (const float*)d_in[i++];      // V x E
  const float* action_embeds = (const float*)d_in[i++];    // ASZ x AD
  const float* relation_embeds = (const float*)d_in[i++];  // ASZ x RD
  const float* stack_Wih = (const float*)d_in[i++];
  const float* stack_Whh = (const float*)d_in[i++];
  const float* stack_bih = (const float*)d_in[i++];
  const float* stack_bhh = (const float*)d_in[i++];
  const float* buffer_Wih = (const float*)d_in[i++];
  const float* buffer_Whh = (const float*)d_in[i++];
  const float* buffer_bih = (const float*)d_in[i++];
  const float* buffer_bhh = (const float*)d_in[i++];
  const float* output_Wih = (const float*)d_in[i++];
  const float* output_Whh = (const float*)d_in[i++];
  const float* output_bih = (const float*)d_in[i++];
  const float* output_bhh = (const float*)d_in[i++];
  const float* action_Wih = (const float*)d_in[i++];
  const float* action_Whh = (const float*)d_in[i++];
  const float* action_bih = (const float*)d_in[i++];
  const float* action_bhh = (const float*)d_in[i++];
  const float* entf_Wih = (const float*)d_in[i++];
  const float* entf_Whh = (const float*)d_in[i++];
  const float* entf_bih = (const float*)d_in[i++];
  const float* entf_bhh = (const float*)d_in[i++];
  const float* entb_Wih = (const float*)d_in[i++];
  const float* entb_Whh = (const float*)d_in[i++];
  const float* entb_bih = (const float*)d_in[i++];
  const float* entb_bhh = (const float*)d_in[i++];
  const float* comb_W = (const float*)d_in[i++];
  const float* comb_b = (const float*)d_in[i++];
  const float* act_W = (const float*)d_in[i++];
  const float* act_b = (const float*)d_in[i++];
  const float* er_W = (const float*)d_in[i++];
  const float* er_b = (const float*)d_in[i++];
  const float* stack_start = (const float*)d_in[i++];
  const float* output_start = (const float*)d_in[i++];
  const float* buffer_start = (const float*)d_in[i++];
  const float* action_start = (const float*)d_in[i++];

  const int L = in_sizes[0];     // 256
  const int nAct = in_sizes[1];  // ~342
  const int T = L + 1;           // 257 buffer rows

  // workspace layout (floats).  buf_in first: the WMMA GEMM's unguarded
  // out-of-range B loads land in the preBuf region right after it.
  float* w = (float*)d_ws;
  float* ws_buf_in = w;          w += (size_t)T * E_DIM;
  float* ws_preBuf = w;          w += (size_t)T * G_DIM;
  float* ws_buf_outs = w;        w += (size_t)T * H_DIM;
  float* ws_stkE = w;            w += (size_t)T * E_DIM;
  float* ws_stkH = w;            w += (size_t)T * H_DIM;
  float* ws_stkC = w;            w += (size_t)T * H_DIM;
  (void)ws_size;
  (void)n_in;
  (void)out_size;

  // Phase 1a: gather buf_in
  k_gather_buf_in<<<T, 128, 0, stream>>>(sentence, word_embeds, buffer_start,
                                         ws_buf_in, L);

  // Phase 1b: preBuf = Wih_buffer @ buf_in^T  (WMMA f32 GEMM, M=2048,K=300,N=257)
  {
    int M = G_DIM, K = E_DIM, N = T;
    int tiles = (M / 16) * ((N + 15) / 16);
    k_wmma_gemm_wxT<<<tiles, 32, 0, stream>>>(buffer_Wih, ws_buf_in, ws_preBuf,
                                              M, K, N);
  }

  // Phase 2: persistent serial parser
  k_parser_seq<<<1, NT, 0, stream>>>(
      actions, nAct, L, ws_preBuf, ws_buf_in, ws_buf_outs, ws_stkE, ws_stkH,
      ws_stkC, stack_Wih, stack_Whh, stack_bih, stack_bhh, buffer_Whh,
      buffer_bih, buffer_bhh, output_Wih, output_Whh, output_bih, output_bhh,
      action_Wih, action_Whh, action_bih, action_bhh, entf_Wih, entf_Whh,
      entf_bih, entf_bhh, entb_Wih, entb_Whh, entb_bih, entb_bhh, comb_W,
      comb_b, act_W, act_b, er_W, er_b, stack_start, output_start,
      action_start, action_embeds, relation_embeds, (float*)d_out);
}